// AdaptiveTopologyLearning_50190987821061
// MI455X (gfx1250) — compile-verified
//
#include <hip/hip_runtime.h>
#include <math.h>

#define Nn   50
#define Sd   2048
#define Hd   4096
#define Cd   64
#define H4d  1024
#define H2d  2048
#define Ed   1225

typedef __attribute__((ext_vector_type(2))) float v2f;
typedef __attribute__((ext_vector_type(8))) float v8f;

__device__ __forceinline__ float gelu_f(float x) {
  return 0.5f * x * (1.0f + erff(x * 0.70710678118654752f));
}
__device__ __forceinline__ float sigm_f(float x) {
  return 1.0f / (1.0f + expf(-x));
}
__device__ __forceinline__ float block_reduce_sum(float v, float* sh) {
  sh[threadIdx.x] = v;
  __syncthreads();
  for (int st = blockDim.x >> 1; st > 0; st >>= 1) {
    if ((int)threadIdx.x < st) sh[threadIdx.x] += sh[threadIdx.x + st];
    __syncthreads();
  }
  float r = sh[0];
  __syncthreads();
  return r;
}

// ---------------------------------------------------------------------------
// Generic fp32 WMMA GEMM:  C[m,n] = act( sum_k A[m,k] * W[n,k] + bias[n] )
//   A: [M,K] row-major, W: [Nout,K] row-major (torch Linear weight)
//   grid = (Nout/256, ceil(M/16)), block = 128 (4 wave32s)
//   each wave: one 16-row x 64-col tile via 4 chained V_WMMA_F32_16X16X4_F32
//   - A row index CLAMPED (not predicated): garbage only reaches D rows the
//     epilogue never stores -> no EXEC-mask branch in the hot loop.
//   - software double-buffering: fragments for k+4 are loaded before the
//     WMMAs for k issue, so s_wait_loadcnt targets loads one iter old.
//   - four independent B base pointers: per-iteration address math is four
//     independent 64-bit adds (dual-issuable) instead of a serial chain.
// ---------------------------------------------------------------------------
__global__ void gemm_f32_wmma(const float* __restrict__ A,
                              const float* __restrict__ W,
                              const float* __restrict__ bias,
                              float* __restrict__ C,
                              int M, int K, int Nout, int act) {
  const int lane = threadIdx.x & 31;
  const int wave = threadIdx.x >> 5;
  const int l16  = lane & 15;
  const int half = lane >> 4;           // 0: K 0..1, 1: K 2..3
  const int rowBase = blockIdx.y * 16;
  const int colBase = blockIdx.x * 256 + wave * 64;

  const int aRow = min(rowBase + l16, M - 1);      // clamped, always in-bounds
  const float* aPtr  = A + (size_t)aRow * K + half * 2;
  const size_t wS    = (size_t)16 * K;
  const float* wPtr0 = W + (size_t)(colBase + l16) * K + half * 2;
  const float* wPtr1 = wPtr0 + wS;
  const float* wPtr2 = wPtr0 + 2 * wS;
  const float* wPtr3 = wPtr0 + 3 * wS;

  v8f acc0 = {}, acc1 = {}, acc2 = {}, acc3 = {};

  v2f a  = *(const v2f*)(aPtr);
  v2f b0 = *(const v2f*)(wPtr0);
  v2f b1 = *(const v2f*)(wPtr1);
  v2f b2 = *(const v2f*)(wPtr2);
  v2f b3 = *(const v2f*)(wPtr3);

#pragma unroll 2
  for (int k = 4; k < K; k += 4) {
    v2f an  = *(const v2f*)(aPtr + k);
    v2f bn0 = *(const v2f*)(wPtr0 + k);
    v2f bn1 = *(const v2f*)(wPtr1 + k);
    v2f bn2 = *(const v2f*)(wPtr2 + k);
    v2f bn3 = *(const v2f*)(wPtr3 + k);
    acc0 = __builtin_amdgcn_wmma_f32_16x16x4_f32(false, a, false, b0, (short)0, acc0, false, false);
    acc1 = __builtin_amdgcn_wmma_f32_16x16x4_f32(false, a, false, b1, (short)0, acc1, false, false);
    acc2 = __builtin_amdgcn_wmma_f32_16x16x4_f32(false, a, false, b2, (short)0, acc2, false, false);
    acc3 = __builtin_amdgcn_wmma_f32_16x16x4_f32(false, a, false, b3, (short)0, acc3, false, false);
    a = an; b0 = bn0; b1 = bn1; b2 = bn2; b3 = bn3;
  }
  acc0 = __builtin_amdgcn_wmma_f32_16x16x4_f32(false, a, false, b0, (short)0, acc0, false, false);
  acc1 = __builtin_amdgcn_wmma_f32_16x16x4_f32(false, a, false, b1, (short)0, acc1, false, false);
  acc2 = __builtin_amdgcn_wmma_f32_16x16x4_f32(false, a, false, b2, (short)0, acc2, false, false);
  acc3 = __builtin_amdgcn_wmma_f32_16x16x4_f32(false, a, false, b3, (short)0, acc3, false, false);

  v8f accs[4] = {acc0, acc1, acc2, acc3};
  for (int t = 0; t < 4; ++t) {
    const int c  = colBase + t * 16 + l16;
    const float bb = bias ? bias[c] : 0.f;
    for (int j = 0; j < 8; ++j) {
      const int r = rowBase + j + half * 8;
      if (r < M) {
        float v = accs[t][j] + bb;
        if (act == 1) v = gelu_f(v);
        C[(size_t)r * Nout + c] = v;
      }
    }
  }
}

// one block per row: y = [gelu]( (x-mean)/sqrt(var+1e-5)*g + b ), L elems
__global__ void layernorm_act(const float* __restrict__ X, float* __restrict__ Y,
                              const float* __restrict__ g, const float* __restrict__ b,
                              int L, int doGelu) {
  __shared__ float sh[256];
  const float* x = X + (size_t)blockIdx.x * L;
  float* y = Y + (size_t)blockIdx.x * L;
  float s = 0.f;
  for (int i = threadIdx.x; i < L; i += 256) s += x[i];
  float mean = block_reduce_sum(s, sh) / (float)L;
  float vs = 0.f;
  for (int i = threadIdx.x; i < L; i += 256) { float d = x[i] - mean; vs += d * d; }
  float var = block_reduce_sum(vs, sh) / (float)L;
  float inv = rsqrtf(var + 1e-5f);
  for (int i = threadIdx.x; i < L; i += 256) {
    float v = (x[i] - mean) * inv * g[i] + b[i];
    y[i] = doGelu ? gelu_f(v) : v;
  }
}

__global__ void chaos_gemv(const float* __restrict__ chaos, const float* __restrict__ W,
                           const float* __restrict__ b, float* __restrict__ out) {
  int h = blockIdx.x * 256 + threadIdx.x;
  if (h < Hd) {
    float s = b[h];
    for (int c = 0; c < Cd; ++c) s += chaos[c] * W[h * Cd + c];
    out[h] = s;
  }
}

// attention over 50 tokens; qkv: [Nn, 3H]; grid (Nn, nheads), block 256
__global__ void attention(const float* __restrict__ qkv, float* __restrict__ out, int nheads) {
  const int i = blockIdx.x, h = blockIdx.y;
  const int d = Hd / nheads;
  __shared__ float sc[64];
  const int wv = threadIdx.x >> 5, ln = threadIdx.x & 31;
  const float scale = rsqrtf((float)d);
  const float* q = qkv + (size_t)i * 3 * Hd + h * d;
  for (int j = wv; j < Nn; j += 8) {
    const float* kk = qkv + (size_t)j * 3 * Hd + Hd + h * d;
    float s = 0.f;
    for (int t = ln; t < d; t += 32) s += q[t] * kk[t];
    for (int off = 16; off > 0; off >>= 1) s += __shfl_down(s, off, 32);
    if (ln == 0) sc[j] = s * scale;
  }
  __syncthreads();
  float mx = -1e30f;
  for (int t = 0; t < Nn; ++t) mx = fmaxf(mx, sc[t]);
  float sum = 0.f;
  for (int t = 0; t < Nn; ++t) sum += expf(sc[t] - mx);
  __syncthreads();
  if ((int)threadIdx.x < Nn) sc[threadIdx.x] = expf(sc[threadIdx.x] - mx) / sum;
  __syncthreads();
  for (int t = threadIdx.x; t < d; t += 256) {
    float o = 0.f;
    for (int j = 0; j < Nn; ++j) o += sc[j] * qkv[(size_t)j * 3 * Hd + 2 * Hd + h * d + t];
    out[(size_t)i * Hd + h * d + t] = o;
  }
}

__global__ void combine3(const float* a, const float* b, const float* c, float* o, int n) {
  int i = blockIdx.x * 256 + threadIdx.x;
  if (i < n) o[i] = (a[i] + b[i] + c[i]) * (1.0f / 3.0f);
}

// grid (Nn, Nn): sim = sigmoid(dot)-eye ; dist = sqrt(max(||a-b||^2,1e-12))
__global__ void pair_stats(const float* __restrict__ comb, float* sim, float* dist) {
  const int i = blockIdx.x, j = blockIdx.y;
  const float* A = comb + (size_t)i * Hd;
  const float* B = comb + (size_t)j * Hd;
  float dp = 0.f, sq = 0.f;
  for (int t = threadIdx.x; t < Hd; t += 256) {
    float x = A[t], y = B[t];
    dp += x * y;
    float d = x - y; sq += d * d;
  }
  __shared__ float sh[256];
  dp = block_reduce_sum(dp, sh);
  sq = block_reduce_sum(sq, sh);
  if (threadIdx.x == 0) {
    sim[i * Nn + j]  = sigm_f(dp) - (i == j ? 1.f : 0.f);
    dist[i * Nn + j] = sqrtf(fmaxf(sq, 1e-12f));
  }
}

__global__ void reduce_max(const float* x, int n, float* out) {
  __shared__ float r[256];
  float m = -1e30f;
  for (int i = threadIdx.x; i < n; i += 256) m = fmaxf(m, x[i]);
  r[threadIdx.x] = m;
  __syncthreads();
  for (int st = 128; st > 0; st >>= 1) {
    if ((int)threadIdx.x < st) r[threadIdx.x] = fmaxf(r[threadIdx.x], r[threadIdx.x + st]);
    __syncthreads();
  }
  if (threadIdx.x == 0) out[0] = r[0];
}

// grid Nn*Nn: dw = sigmoid( relu(nd*W1+b1) . W2 + b2 ) * (1-eye)
__global__ void dist_mlp(const float* dist, const float* dmax,
                         const float* W1, const float* b1,
                         const float* W2, const float* b2, float* dadj) {
  const int p = blockIdx.x, i = p / Nn, j = p % Nn;
  const float nd = dist[p] / (dmax[0] + 1e-8f);
  float s = 0.f;
  for (int t = threadIdx.x; t < H4d; t += 256)
    s += fmaxf(nd * W1[t] + b1[t], 0.f) * W2[t];
  __shared__ float sh[256];
  s = block_reduce_sum(s, sh);
  if (threadIdx.x == 0) dadj[p] = (i == j) ? 0.f : sigm_f(s + b2[0]);
}

__global__ void build_proj_input(const float* comb, const float* enc_c, float* pin) {
  int idx = blockIdx.x * 256 + threadIdx.x;
  if (idx < Nn * 2 * Hd) {
    int i = idx / (2 * Hd), c = idx % (2 * Hd);
    pin[idx] = (c < Hd) ? comb[(size_t)i * Hd + c] : enc_c[c - Hd];
  }
}

__global__ void gelu_inplace(float* x, int n) {
  int i = blockIdx.x * 256 + threadIdx.x;
  if (i < n) x[i] = gelu_f(x[i]);
}

// grid (Nn, Nn): cadj = sigmoid(dot(gproj_i, W_pr_j)+b_pr_j) * (1-eye)
__global__ void chaos_adj_kernel(const float* __restrict__ gproj,
                                 const float* __restrict__ W_pr,
                                 const float* __restrict__ b_pr, float* cadj) {
  const int i = blockIdx.x, j = blockIdx.y;
  const float* A = gproj + (size_t)i * Hd;
  const float* W = W_pr + (size_t)j * Hd;
  float s = 0.f;
  for (int t = threadIdx.x; t < Hd; t += 256) s += A[t] * W[t];
  __shared__ float sh[256];
  s = block_reduce_sum(s, sh);
  if (threadIdx.x == 0) cadj[i * Nn + j] = (i == j) ? 0.f : sigm_f(s + b_pr[j]);
}

__global__ void combine_adj(const float* sim, const float* dadj, const float* cadj, float* out) {
  int idx = blockIdx.x * 256 + threadIdx.x;
  if (idx < Nn * Nn) {
    int i = idx / Nn, j = idx % Nn;
    float st = (i == j) ? 0.f : 1.f;
    out[idx] = 0.3f * st + 0.25f * sim[idx] + 0.25f * dadj[idx] + 0.2f * cadj[idx];
  }
}

__global__ void build_edge_idx(int* ei, int* ej) {
  int e = blockIdx.x * 256 + threadIdx.x;
  if (e < Ed) {
    int i = 0, rem = e;
    while (rem >= Nn - 1 - i) { rem -= (Nn - 1 - i); ++i; }
    ei[e] = i;
    ej[e] = i + 1 + rem;
  }
}

// grid (ceil(8256/256), Ed)
__global__ void build_edge_input(const float* __restrict__ comb, const float* __restrict__ chaos,
                                 const int* ei, const int* ej, float* __restrict__ ein) {
  const int e = blockIdx.y;
  const int col = blockIdx.x * 256 + threadIdx.x;
  const int Ke = 2 * Hd + Cd;
  if (col < Ke) {
    const int i = ei[e], j = ej[e];
    float v;
    if (col < Hd)          v = comb[(size_t)i * Hd + col];
    else if (col < 2 * Hd) v = comb[(size_t)j * Hd + (col - Hd)];
    else                   v = chaos[col - 2 * Hd];
    ein[(size_t)e * Ke + col] = v;
  }
}

__global__ void edge_importance(const float* __restrict__ h2, const float* __restrict__ W3,
                                const float* __restrict__ b3, float* imp) {
  const int e = blockIdx.x;
  const float* x = h2 + (size_t)e * H2d;
  float s = 0.f;
  for (int t = threadIdx.x; t < H2d; t += 256) s += x[t] * W3[t];
  __shared__ float sh[256];
  s = block_reduce_sum(s, sh);
  if (threadIdx.x == 0) imp[e] = sigm_f(s + b3[0]);
}

// grid Nn, block 64: es row build + softmax(es/temp) + fa0 = combined_adj * es_soft
__global__ void edge_softmax(const float* __restrict__ cadj, const float* __restrict__ imp,
                             const float* __restrict__ edge_temp, float* __restrict__ fa0) {
  const int i = blockIdx.x, j = threadIdx.x;
  __shared__ float sv[64];
  float esv = 1.0f;
  if (j < Nn && j != i) {
    const int a = min(i, j), b = max(i, j);
    const int e = a * (Nn - 1) - (a * (a - 1)) / 2 + (b - a - 1);
    esv = (cadj[a * Nn + b] > 0.1f) ? imp[e] : 1.0f;
  }
  const float temp = sigm_f(edge_temp[0]) + 0.1f;
  sv[j] = (j < Nn) ? esv / temp : -1e30f;
  __syncthreads();
  float mx = -1e30f;
  for (int t = 0; t < Nn; ++t) mx = fmaxf(mx, sv[t]);
  float sum = 0.f;
  for (int t = 0; t < Nn; ++t) sum += expf(sv[t] - mx);
  if (j < Nn) fa0[i * Nn + j] = cadj[i * Nn + j] * (expf(sv[j] - mx) / sum);
}

// grid Nn, block 64: fa = (fa0+fa0^T)/2 + 0.5*eye ; out = fa / (rowsum+1e-8)
__global__ void finalize(const float* __restrict__ fa0, float* __restrict__ out) {
  const int i = blockIdx.x, j = threadIdx.x;
  __shared__ float sv[64];
  float v = 0.f;
  if (j < Nn) {
    v = 0.5f * (fa0[i * Nn + j] + fa0[j * Nn + i]);
    if (i == j) v += 0.5f;
  }
  sv[j] = (j < Nn) ? v : 0.f;
  __syncthreads();
  float s = 0.f;
  for (int t = 0; t < Nn; ++t) s += sv[t];
  if (j < Nn) out[i * Nn + j] = v / (s + 1e-8f);
}

extern "C" void kernel_launch(void* const* d_in, const int* in_sizes, int n_in,
                              void* d_out, int out_size, void* d_ws, size_t ws_size,
                              hipStream_t stream) {
  (void)in_sizes; (void)n_in; (void)out_size; (void)ws_size;
  const float* node    = (const float*)d_in[0];
  const float* chaos   = (const float*)d_in[1];
  const float* W_ne    = (const float*)d_in[2];
  const float* b_ne    = (const float*)d_in[3];
  const float* g_ne    = (const float*)d_in[4];
  const float* be_ne   = (const float*)d_in[5];
  const float* W_ce    = (const float*)d_in[6];
  const float* b_ce    = (const float*)d_in[7];
  const float* g_ce    = (const float*)d_in[8];
  const float* be_ce   = (const float*)d_in[9];
  const float* Wqkv_l  = (const float*)d_in[10];
  const float* bqkv_l  = (const float*)d_in[11];
  const float* Wo_l    = (const float*)d_in[12];
  const float* bo_l    = (const float*)d_in[13];
  const float* Wqkv_g  = (const float*)d_in[14];
  const float* bqkv_g  = (const float*)d_in[15];
  const float* Wo_g    = (const float*)d_in[16];
  const float* bo_g    = (const float*)d_in[17];
  const float* W_d1    = (const float*)d_in[18];
  const float* b_d1    = (const float*)d_in[19];
  const float* W_d2    = (const float*)d_in[20];
  const float* b_d2    = (const float*)d_in[21];
  const float* W_ap    = (const float*)d_in[22];
  const float* b_ap    = (const float*)d_in[23];
  const float* W_pr    = (const float*)d_in[24];
  const float* b_pr    = (const float*)d_in[25];
  const float* W_e1    = (const float*)d_in[26];
  const float* b_e1    = (const float*)d_in[27];
  const float* g_e     = (const float*)d_in[28];
  const float* be_e    = (const float*)d_in[29];
  const float* W_e2    = (const float*)d_in[30];
  const float* b_e2    = (const float*)d_in[31];
  const float* W_e3    = (const float*)d_in[32];
  const float* b_e3    = (const float*)d_in[33];
  const float* e_temp  = (const float*)d_in[34];

  char* ws = (char*)d_ws;
  size_t off = 0;
  auto alloc = [&](size_t nfloats) -> float* {
    float* p = (float*)(ws + off);
    off += ((nfloats * sizeof(float) + 255) / 256) * 256;
    return p;
  };
  float* enc_n = alloc((size_t)Nn * Hd);
  float* enc_c = alloc(Hd);
  float* qkv   = alloc((size_t)Nn * 3 * Hd);   // reused local then global
  float* att   = alloc((size_t)Nn * Hd);       // reused local then global
  float* loc   = alloc((size_t)Nn * Hd);
  float* glo   = alloc((size_t)Nn * Hd);
  float* comb  = alloc((size_t)Nn * Hd);
  float* sim   = alloc(Nn * Nn);
  float* dist  = alloc(Nn * Nn);
  float* dmax  = alloc(8);
  float* dadj  = alloc(Nn * Nn);
  float* pin   = alloc((size_t)Nn * 2 * Hd);
  float* proj  = alloc((size_t)Nn * Hd);
  float* cadj  = alloc(Nn * Nn);
  float* comA  = alloc(Nn * Nn);
  float* fa0   = alloc(Nn * Nn);
  float* imp   = alloc(1280);
  int*   ei    = (int*)alloc(1280);
  int*   ej    = (int*)alloc(1280);
  float* ein   = alloc((size_t)Ed * (2 * Hd + Cd));
  float* h1    = alloc((size_t)Ed * Hd);
  float* h2    = alloc((size_t)Ed * H2d);

  const int Ke = 2 * Hd + Cd;          // 8256
  const int rowT_n = (Nn + 15) / 16;   // 4
  const int rowT_e = (Ed + 15) / 16;   // 77

  // node encoder: GEMM + LN + GELU
  gemm_f32_wmma<<<dim3(Hd / 256, rowT_n), 128, 0, stream>>>(node, W_ne, b_ne, enc_n, Nn, Sd, Hd, 0);
  layernorm_act<<<Nn, 256, 0, stream>>>(enc_n, enc_n, g_ne, be_ne, Hd, 1);
  // chaos encoder
  chaos_gemv<<<Hd / 256, 256, 0, stream>>>(chaos, W_ce, b_ce, enc_c);
  layernorm_act<<<1, 256, 0, stream>>>(enc_c, enc_c, g_ce, be_ce, Hd, 1);
  // local MHA (4 heads)
  gemm_f32_wmma<<<dim3(3 * Hd / 256, rowT_n), 128, 0, stream>>>(enc_n, Wqkv_l, bqkv_l, qkv, Nn, Hd, 3 * Hd, 0);
  attention<<<dim3(Nn, 4), 256, 0, stream>>>(qkv, att, 4);
  gemm_f32_wmma<<<dim3(Hd / 256, rowT_n), 128, 0, stream>>>(att, Wo_l, bo_l, loc, Nn, Hd, Hd, 0);
  // global MHA (2 heads)
  gemm_f32_wmma<<<dim3(3 * Hd / 256, rowT_n), 128, 0, stream>>>(enc_n, Wqkv_g, bqkv_g, qkv, Nn, Hd, 3 * Hd, 0);
  attention<<<dim3(Nn, 2), 256, 0, stream>>>(qkv, att, 2);
  gemm_f32_wmma<<<dim3(Hd / 256, rowT_n), 128, 0, stream>>>(att, Wo_g, bo_g, glo, Nn, Hd, Hd, 0);
  // combined nodes
  combine3<<<(Nn * Hd + 255) / 256, 256, 0, stream>>>(loc, glo, enc_n, comb, Nn * Hd);
  // similarity + distance adjacency
  pair_stats<<<dim3(Nn, Nn), 256, 0, stream>>>(comb, sim, dist);
  reduce_max<<<1, 256, 0, stream>>>(dist, Nn * Nn, dmax);
  dist_mlp<<<Nn * Nn, 256, 0, stream>>>(dist, dmax, W_d1, b_d1, W_d2, b_d2, dadj);
  // chaos adjacency
  build_proj_input<<<(Nn * 2 * Hd + 255) / 256, 256, 0, stream>>>(comb, enc_c, pin);
  gemm_f32_wmma<<<dim3(Hd / 256, rowT_n), 128, 0, stream>>>(pin, W_ap, b_ap, proj, Nn, 2 * Hd, Hd, 0);
  gelu_inplace<<<(Nn * Hd + 255) / 256, 256, 0, stream>>>(proj, Nn * Hd);
  chaos_adj_kernel<<<dim3(Nn, Nn), 256, 0, stream>>>(proj, W_pr, b_pr, cadj);
  combine_adj<<<(Nn * Nn + 255) / 256, 256, 0, stream>>>(sim, dadj, cadj, comA);
  // edge MLP (dominant GEMMs)
  build_edge_idx<<<(Ed + 255) / 256, 256, 0, stream>>>(ei, ej);
  build_edge_input<<<dim3((Ke + 255) / 256, Ed), 256, 0, stream>>>(comb, chaos, ei, ej, ein);
  gemm_f32_wmma<<<dim3(Hd / 256, rowT_e), 128, 0, stream>>>(ein, W_e1, b_e1, h1, Ed, Ke, Hd, 0);
  layernorm_act<<<Ed, 256, 0, stream>>>(h1, h1, g_e, be_e, Hd, 1);
  gemm_f32_wmma<<<dim3(H2d / 256, rowT_e), 128, 0, stream>>>(h1, W_e2, b_e2, h2, Ed, Hd, H2d, 1);
  edge_importance<<<Ed, 256, 0, stream>>>(h2, W_e3, b_e3, imp);
  // edge strength softmax + final adjacency
  edge_softmax<<<Nn, 64, 0, stream>>>(comA, imp, e_temp, fa0);
  finalize<<<Nn, 64, 0, stream>>>(fa0, (float*)d_out);
}